// LinearQNet_55637006352517
// MI455X (gfx1250) — compile-verified
//
#include <hip/hip_runtime.h>

// ---------------------------------------------------------------------------
// LinearQNet forward: out = (sigmoid(sigmoid(x@W1^T)@W2^T))@W5^T
//   x: [B,20] f32, W1: [6,20], W2: [6,6], W5: [6,6], out: [B,6]
// Memory-bound (AI ~3.7 flop/B, floor ~19us @ 23.3 TB/s). Strategy: stream
// 16-row tiles per wave, compute transposed  Y^T = W5*sig(W2*sig(W1*X^T))
// with V_WMMA_F32_16X16X4_F32 so weights live in the A operand and each
// layer's D layout converts to the next layer's B operand with one
// ds_bpermute per hidden row (12/tile). Every input/output byte is touched
// exactly once, so the streaming x loads and out stores use non-temporal
// hints (TH_NT) to avoid rinsing the 436 MB stream through L2; the weight
// loads stay regular-temporal since every wave reuses them.
// ---------------------------------------------------------------------------

typedef __attribute__((ext_vector_type(2))) float v2f;
typedef __attribute__((ext_vector_type(8))) float v8f;

__device__ __forceinline__ v2f ld2(const float* p) {
    return *reinterpret_cast<const v2f*>(p);
}

// streaming (non-temporal) float2 load: global_load_b64 th:TH_LOAD_NT
__device__ __forceinline__ v2f ld2_nt(const float* p) {
    return __builtin_nontemporal_load(reinterpret_cast<const v2f*>(p));
}

__device__ __forceinline__ void st2_nt(float* p, v2f v) {
    __builtin_nontemporal_store(v, reinterpret_cast<v2f*>(p));
}

__device__ __forceinline__ float sigmoid_fast(float v) {
    // 1 / (1 + 2^(-x * log2 e))  -> v_exp_f32 + v_add + v_rcp
    float e = __builtin_amdgcn_exp2f(v * -1.44269504088896341f);
    return __builtin_amdgcn_rcpf(1.0f + e);
}

// broadcast value held by lane (lane&15) to both wave halves
__device__ __forceinline__ float bperm16(float v, int lane15x4) {
    return __int_as_float(
        __builtin_amdgcn_ds_bpermute(lane15x4, __float_as_int(v)));
}

__global__ void __launch_bounds__(256)
mlp_qnet_kernel(const float* __restrict__ x,
                const float* __restrict__ W1,
                const float* __restrict__ W2,
                const float* __restrict__ W5,
                float* __restrict__ out,
                int ntiles) {
    const int lane  = threadIdx.x & 31;
    const int m     = lane & 15;          // A-matrix row / output column lane
    const int koff  = (lane >> 4) << 1;   // 0 for lanes 0-15, 2 for 16-31
    const bool lo   = lane < 16;
    const int bidx4 = m << 2;             // byte index for ds_bpermute
    const v2f z2    = {0.0f, 0.0f};

    // ---- per-lane weight preload (A operands, zero-padded to 16 rows) ----
    // Layer 1: A = W1 (6x20 -> 16x20), chunk c covers K = 4c .. 4c+3
    v2f a1[5];
#pragma unroll
    for (int c = 0; c < 5; ++c)
        a1[c] = (m < 6) ? ld2(W1 + m * 20 + c * 4 + koff) : z2;
    // Layers 2/3: A = W2 / W5 (6x6 -> 16x8), chunk0 K=0..3, chunk1 K=4..7
    v2f a2[2], a3[2];
    a2[0] = (m < 6) ? ld2(W2 + m * 6 + koff) : z2;          // K = koff,koff+1
    a2[1] = (m < 6 && lo) ? ld2(W2 + m * 6 + 4) : z2;       // K = 4,5 (6,7 -> 0)
    a3[0] = (m < 6) ? ld2(W5 + m * 6 + koff) : z2;
    a3[1] = (m < 6 && lo) ? ld2(W5 + m * 6 + 4) : z2;

    const int wave   = blockIdx.x * (blockDim.x >> 5) + (threadIdx.x >> 5);
    const int nwaves = gridDim.x * (blockDim.x >> 5);

    for (int t = wave; t < ntiles; t += nwaves) {
        const long base = (long)t * 16;
        // B operand of layer 1 = X^T tile:
        //   VGPR r, lane -> X[base + (lane&15)][4c + koff + r]
        const float* xrow = x + (base + m) * 20 + koff;

        if (t + nwaves < ntiles) {  // speculative prefetch of next tile's rows
            __builtin_prefetch(x + (base + (long)nwaves * 16 + m) * 20, 0, 0);
        }

        v2f b0 = ld2_nt(xrow);
        v2f b1 = ld2_nt(xrow + 4);
        v2f b2 = ld2_nt(xrow + 8);
        v2f b3 = ld2_nt(xrow + 12);
        v2f b4 = ld2_nt(xrow + 16);

        // ---- layer 1: G1 = W1 @ X^T   (16x16, rows 0-5 valid) ----
        v8f acc = {0.f, 0.f, 0.f, 0.f, 0.f, 0.f, 0.f, 0.f};
        acc = __builtin_amdgcn_wmma_f32_16x16x4_f32(false, a1[0], false, b0, (short)0, acc, false, false);
        acc = __builtin_amdgcn_wmma_f32_16x16x4_f32(false, a1[1], false, b1, (short)0, acc, false, false);
        acc = __builtin_amdgcn_wmma_f32_16x16x4_f32(false, a1[2], false, b2, (short)0, acc, false, false);
        acc = __builtin_amdgcn_wmma_f32_16x16x4_f32(false, a1[3], false, b3, (short)0, acc, false, false);
        acc = __builtin_amdgcn_wmma_f32_16x16x4_f32(false, a1[4], false, b4, (short)0, acc, false, false);

        // ---- sigmoid + redistribute D-layout rows 0..5 into B-layout ----
        float h0 = bperm16(sigmoid_fast(acc[0]), bidx4);
        float h1 = bperm16(sigmoid_fast(acc[1]), bidx4);
        float h2 = bperm16(sigmoid_fast(acc[2]), bidx4);
        float h3 = bperm16(sigmoid_fast(acc[3]), bidx4);
        float h4 = bperm16(sigmoid_fast(acc[4]), bidx4);
        float h5 = bperm16(sigmoid_fast(acc[5]), bidx4);
        v2f bl2c0 = {lo ? h0 : h2, lo ? h1 : h3};     // K = 0/2, 1/3
        v2f bl2c1 = {lo ? h4 : 0.0f, lo ? h5 : 0.0f}; // K = 4/6(pad), 5/7(pad)

        // ---- layer 2: G2 = W2 @ H1 ----
        v8f acc2 = {0.f, 0.f, 0.f, 0.f, 0.f, 0.f, 0.f, 0.f};
        acc2 = __builtin_amdgcn_wmma_f32_16x16x4_f32(false, a2[0], false, bl2c0, (short)0, acc2, false, false);
        acc2 = __builtin_amdgcn_wmma_f32_16x16x4_f32(false, a2[1], false, bl2c1, (short)0, acc2, false, false);

        float g0 = bperm16(sigmoid_fast(acc2[0]), bidx4);
        float g1 = bperm16(sigmoid_fast(acc2[1]), bidx4);
        float g2 = bperm16(sigmoid_fast(acc2[2]), bidx4);
        float g3 = bperm16(sigmoid_fast(acc2[3]), bidx4);
        float g4 = bperm16(sigmoid_fast(acc2[4]), bidx4);
        float g5 = bperm16(sigmoid_fast(acc2[5]), bidx4);
        v2f bl3c0 = {lo ? g0 : g2, lo ? g1 : g3};
        v2f bl3c1 = {lo ? g4 : 0.0f, lo ? g5 : 0.0f};

        // ---- layer 3: Y^T = W5 @ H2 ----
        v8f acc3 = {0.f, 0.f, 0.f, 0.f, 0.f, 0.f, 0.f, 0.f};
        acc3 = __builtin_amdgcn_wmma_f32_16x16x4_f32(false, a3[0], false, bl3c0, (short)0, acc3, false, false);
        acc3 = __builtin_amdgcn_wmma_f32_16x16x4_f32(false, a3[1], false, bl3c1, (short)0, acc3, false, false);

        // ---- store: lane n (<16) owns out row base+n, VGPRs 0..5 = cols ----
        if (lo) {
            float* o = out + (base + m) * 6;
            v2f s0 = {acc3[0], acc3[1]};
            v2f s1 = {acc3[2], acc3[3]};
            v2f s2 = {acc3[4], acc3[5]};
            st2_nt(o,     s0);
            st2_nt(o + 2, s1);
            st2_nt(o + 4, s2);
        }
    }
}

extern "C" void kernel_launch(void* const* d_in, const int* in_sizes, int n_in,
                              void* d_out, int out_size, void* d_ws, size_t ws_size,
                              hipStream_t stream) {
    const float* x  = (const float*)d_in[0];
    const float* W1 = (const float*)d_in[1];
    const float* W2 = (const float*)d_in[2];
    const float* W5 = (const float*)d_in[3];
    float* out = (float*)d_out;

    const long B      = (long)in_sizes[0] / 20;   // rows (4,194,304)
    const int  ntiles = (int)(B / 16);            // 16-row tiles

    const int threads       = 256;                // 8 wave32
    const int wavesPerBlock = threads / 32;
    const int tilesPerWave  = 4;                  // amortize weight preload
    int totalWaves = (ntiles + tilesPerWave - 1) / tilesPerWave;
    int blocks     = (totalWaves + wavesPerBlock - 1) / wavesPerBlock;
    if (blocks < 1) blocks = 1;

    mlp_qnet_kernel<<<blocks, threads, 0, stream>>>(x, W1, W2, W5, out, ntiles);
}